// RelayTokenGeometricConsistencyReranker_43757126811715
// MI455X (gfx1250) — compile-verified
//
#include <hip/hip_runtime.h>
#include <hip/hip_bf16.h>
#include <math.h>

#define BB 64
#define NN 1024
#define CC 256
#define KK 512
#define TT 64

typedef __attribute__((ext_vector_type(2))) float v2f;
typedef __attribute__((ext_vector_type(8))) float v8f;

// CDNA5 f32 matrix op: D(16x16,f32) = A(16x4,f32) x B(4x16,f32) + C
// A: lanes 0-15 hold rows M=0..15 with K=k0,k0+1 ; lanes 16-31 hold K=k0+2,k0+3
// B: lanes 0-15 hold cols N=0..15 with K rows k0,k0+1 ; lanes 16-31 K=k0+2,k0+3
// C/D: VGPR r -> row r (lanes 0-15) / row r+8 (lanes 16-31), col = lane%16
__device__ __forceinline__ v8f wmma4(v2f a, v2f b, v8f c) {
  return __builtin_amdgcn_wmma_f32_16x16x4_f32(false, a, false, b, (short)0, c,
                                               false, false);
}

// ---------------------------------------------------------------------------
// Kernel 1: per-batch masked top-K (descending, lower-index tie-break) via
// bitonic sort of (orderable-key, ~index) packed u64, plus centroid gather
// with conditional shift/scale.
// ---------------------------------------------------------------------------
__global__ void topk_kernel(const float* __restrict__ attn,
                            const float* __restrict__ cent,
                            const int* __restrict__ num_rt,
                            const float* __restrict__ sas,
                            int* __restrict__ topIdx,
                            float* __restrict__ centK) {
  int b = blockIdx.x;
  int tid = threadIdx.x;  // 512 threads
  __shared__ unsigned long long key[NN];
  int nr = num_rt[b];
  for (int i = tid; i < NN; i += 512) {
    float a = (i < nr) ? attn[b * NN + i] : -1e9f;
    unsigned ua = __float_as_uint(a);
    ua = (ua & 0x80000000u) ? ~ua : (ua | 0x80000000u);  // monotone map
    key[i] = ((unsigned long long)ua << 32) |
             (unsigned long long)(0xFFFFFFFFu - (unsigned)i);
  }
  __syncthreads();
  for (int ksz = 2; ksz <= NN; ksz <<= 1) {
    for (int j = ksz >> 1; j > 0; j >>= 1) {
      for (int i = tid; i < NN; i += 512) {
        int ixj = i ^ j;
        if (ixj > i) {
          unsigned long long A = key[i], Bv = key[ixj];
          bool up = ((i & ksz) == 0);
          bool sw = up ? (A < Bv) : (A > Bv);  // descending sort
          if (sw) { key[i] = Bv; key[ixj] = A; }
        }
      }
      __syncthreads();
    }
  }
  // top 512 live in key[0..511]
  unsigned idx = 0xFFFFFFFFu - (unsigned)(key[tid] & 0xFFFFFFFFull);
  topIdx[b * KK + tid] = (int)idx;
  float sh0 = sas[b * 4 + 0], sh1 = sas[b * 4 + 1], sh2 = sas[b * 4 + 2];
  float sc = sas[b * 4 + 3];
  bool valid = (int)idx < nr;
  float c0 = cent[((size_t)b * NN + idx) * 3 + 0];
  float c1 = cent[((size_t)b * NN + idx) * 3 + 1];
  float c2 = cent[((size_t)b * NN + idx) * 3 + 2];
  size_t o = ((size_t)b * KK + tid) * 3;
  centK[o + 0] = valid ? c0 * sc + sh0 : c0;
  centK[o + 1] = valid ? c1 * sc + sh1 : c1;
  centK[o + 2] = valid ? c2 * sc + sh2 : c2;
}

// ---------------------------------------------------------------------------
// Kernel 2: proj[b] = rt_k[b] @ W_in + b_in  (per batch 512x256 @ 256x256),
// WMMA f32 16x16x4, rows gathered through topIdx.
// grid (KK/16, CC/128, BB), block 256 (8 waves, wave -> one 16-col tile).
// ---------------------------------------------------------------------------
__global__ void proj_gemm_kernel(const float* __restrict__ rt,
                                 const float* __restrict__ Win,
                                 const float* __restrict__ bin,
                                 const int* __restrict__ topIdx,
                                 float* __restrict__ proj) {
  int b = blockIdx.z;
  int lane = threadIdx.x & 31, wave = threadIdx.x >> 5;
  int rowTile = blockIdx.x;
  int colBase = blockIdx.y * 128 + wave * 16;
  int l15 = lane & 15;
  int kOff = (lane >> 4) * 2;
  int srcRow = topIdx[b * KK + rowTile * 16 + l15];
  const float* Arow = rt + ((size_t)b * NN + srcRow) * CC;
  int ncol = colBase + l15;
  v8f acc = {};
  for (int k0 = 0; k0 < CC; k0 += 4) {
    v2f a = *(const v2f*)(Arow + k0 + kOff);
    v2f bm;
    bm[0] = Win[(size_t)(k0 + kOff) * CC + ncol];
    bm[1] = Win[(size_t)(k0 + kOff + 1) * CC + ncol];
    acc = wmma4(a, bm, acc);
  }
  float bias = bin[ncol];
  int rbase = rowTile * 16 + (lane >> 4) * 8;
#pragma unroll
  for (int r = 0; r < 8; ++r)
    proj[((size_t)b * KK + rbase + r) * CC + ncol] = acc[r] + bias;
}

// ---------------------------------------------------------------------------
// Kernel 3: row squared-norms of proj (one wave per row).
// ---------------------------------------------------------------------------
__global__ void rownorm_kernel(const float* __restrict__ proj,
                               float* __restrict__ n2) {
  int gw = blockIdx.x * 8 + (threadIdx.x >> 5);
  int lane = threadIdx.x & 31;
  const float* p = proj + (size_t)gw * CC;
  float s = 0.f;
  for (int c = lane; c < CC; c += 32) { float x = p[c]; s += x * x; }
  for (int m = 16; m >= 1; m >>= 1) s += __shfl_xor(s, m, 32);
  if (lane == 0) n2[gw] = s;
}

// ---------------------------------------------------------------------------
// Kernel 4: fd[k,l] = |a_k|^2 + |b_l|^2 - 2 a_k.b_l via WMMA (K=256 depth),
// fused row-wise argmin (jnp.argmin tie-break = lowest l).
// grid (KK/16 rowTiles, 2, TT), block 256; wave w owns cols [w*64, w*64+64).
// ---------------------------------------------------------------------------
__global__ void fd_argmin_kernel(const float* __restrict__ proj,
                                 const float* __restrict__ n2,
                                 const int* __restrict__ anc,
                                 const int* __restrict__ pos,
                                 const int* __restrict__ neg,
                                 int* __restrict__ corr) {
  int t = blockIdx.z, nn = blockIdx.y, rowTile = blockIdx.x;
  int lane = threadIdx.x & 31, wave = threadIdx.x >> 5;
  int l15 = lane & 15;
  int kOff = (lane >> 4) * 2;
  int ancB = anc[t];
  int nbrB = (nn == 0) ? pos[t] : neg[t];
  const float* Arow = proj + ((size_t)ancB * KK + rowTile * 16 + l15) * CC;
  int colBase = wave * 64;
  const float* Bbase = proj + (size_t)nbrB * KK * CC;
  const float* Bp[4];
#pragma unroll
  for (int ct = 0; ct < 4; ++ct)
    Bp[ct] = Bbase + (size_t)(colBase + ct * 16 + l15) * CC + kOff;
  v8f acc[4] = {{}, {}, {}, {}};
  for (int k0 = 0; k0 < CC; k0 += 4) {
    v2f a = *(const v2f*)(Arow + k0 + kOff);
#pragma unroll
    for (int ct = 0; ct < 4; ++ct) {
      v2f bm = *(const v2f*)(Bp[ct] + k0);
      acc[ct] = wmma4(a, bm, acc[ct]);
    }
  }
  int rbase = rowTile * 16 + (lane >> 4) * 8;
  float rowN[8];
#pragma unroll
  for (int r = 0; r < 8; ++r) rowN[r] = n2[ancB * KK + rbase + r];
  float bestV[8]; int bestI[8];
#pragma unroll
  for (int r = 0; r < 8; ++r) { bestV[r] = 3.4e38f; bestI[r] = 0x7fffffff; }
#pragma unroll
  for (int ct = 0; ct < 4; ++ct) {
    int cn = colBase + ct * 16 + l15;
    float cN = n2[nbrB * KK + cn];
#pragma unroll
    for (int r = 0; r < 8; ++r) {
      float fd = rowN[r] + cN - 2.0f * acc[ct][r];
      if (fd < bestV[r] || (fd == bestV[r] && cn < bestI[r])) {
        bestV[r] = fd; bestI[r] = cn;
      }
    }
  }
  // reduce across the 16 lanes of each half-wave row group
#pragma unroll
  for (int r = 0; r < 8; ++r) {
    for (int m = 1; m < 16; m <<= 1) {
      float ov = __shfl_xor(bestV[r], m, 32);
      int oi = __shfl_xor(bestI[r], m, 32);
      if (ov < bestV[r] || (ov == bestV[r] && oi < bestI[r])) {
        bestV[r] = ov; bestI[r] = oi;
      }
    }
  }
  __shared__ float sMin[8][2][8];
  __shared__ int sIdx[8][2][8];
  if (l15 == 0) {
    int g = lane >> 4;
#pragma unroll
    for (int r = 0; r < 8; ++r) { sMin[r][g][wave] = bestV[r]; sIdx[r][g][wave] = bestI[r]; }
  }
  __syncthreads();
  if (threadIdx.x < 16) {
    int r = threadIdx.x & 7, g = threadIdx.x >> 3;
    float bv = sMin[r][g][0]; int bi = sIdx[r][g][0];
    for (int w = 1; w < 8; ++w) {
      float ov = sMin[r][g][w]; int oi = sIdx[r][g][w];
      if (ov < bv || (ov == bv && oi < bi)) { bv = ov; bi = oi; }
    }
    corr[((size_t)(t * 2 + nn)) * KK + rowTile * 16 + g * 8 + r] = bi;
  }
}

// ---------------------------------------------------------------------------
// Kernel 5: s = relu(cos(anc_f_k, nn_f_corr)) and gather q_c. Wave per row.
// ---------------------------------------------------------------------------
__global__ void s_qc_kernel(const float* __restrict__ proj,
                            const float* __restrict__ n2,
                            const float* __restrict__ centK,
                            const int* __restrict__ corr,
                            const int* __restrict__ anc,
                            const int* __restrict__ pos,
                            const int* __restrict__ neg,
                            float* __restrict__ sbuf,
                            float* __restrict__ qcb) {
  int gw = blockIdx.x * 8 + (threadIdx.x >> 5);
  int lane = threadIdx.x & 31;
  int t = gw >> 10;
  int rest = gw & 1023;
  int nn = rest >> 9;
  int k = rest & 511;
  int ancB = anc[t];
  int nbrB = (nn == 0) ? pos[t] : neg[t];
  int tn = t * 2 + nn;
  int l = corr[(size_t)tn * KK + k];
  const float* fa = proj + ((size_t)ancB * KK + k) * CC;
  const float* fq = proj + ((size_t)nbrB * KK + l) * CC;
  float dot = 0.f;
  for (int c = lane; c < CC; c += 32) dot += fa[c] * fq[c];
  for (int m = 16; m >= 1; m >>= 1) dot += __shfl_xor(dot, m, 32);
  if (lane == 0) {
    float na = n2[ancB * KK + k], nq = n2[nbrB * KK + l];
    float cosv = dot / (sqrtf(na) * sqrtf(nq) + 1e-8f);
    sbuf[(size_t)tn * KK + k] = fmaxf(cosv, 0.f);
  }
  if (lane < 3)
    qcb[((size_t)tn * KK + k) * 3 + lane] = centK[((size_t)nbrB * KK + l) * 3 + lane];
}

// ---------------------------------------------------------------------------
// Kernel 6: geometric-consistency bitmask geo[k][l] per (t,n). 128 blocks x512.
// ---------------------------------------------------------------------------
__global__ void geo_kernel(const float* __restrict__ centK,
                           const float* __restrict__ qcb,
                           const int* __restrict__ num_rt,
                           const int* __restrict__ anc,
                           const int* __restrict__ pos,
                           const int* __restrict__ neg,
                           unsigned* __restrict__ geo) {
  int tn = blockIdx.x;
  int t = tn >> 1, nn = tn & 1;
  int k = threadIdx.x;
  int ancB = anc[t];
  int nbrB = (nn == 0) ? pos[t] : neg[t];
  int nrA = num_rt[ancB]; int neA = nrA < KK ? nrA : KK;
  int nrN = num_rt[nbrB]; int neN = nrN < KK ? nrN : KK;
  __shared__ float pc[KK][3];
  __shared__ float qc[KK][3];
  __shared__ float npn[KK];
  __shared__ float nqn[KK];
  float p0 = centK[((size_t)ancB * KK + k) * 3 + 0];
  float p1 = centK[((size_t)ancB * KK + k) * 3 + 1];
  float p2 = centK[((size_t)ancB * KK + k) * 3 + 2];
  float q0 = qcb[((size_t)tn * KK + k) * 3 + 0];
  float q1 = qcb[((size_t)tn * KK + k) * 3 + 1];
  float q2 = qcb[((size_t)tn * KK + k) * 3 + 2];
  pc[k][0] = p0; pc[k][1] = p1; pc[k][2] = p2;
  qc[k][0] = q0; qc[k][1] = q1; qc[k][2] = q2;
  float npk = p0 * p0 + p1 * p1 + p2 * p2;
  float nqk = q0 * q0 + q1 * q1 + q2 * q2;
  npn[k] = npk; nqn[k] = nqk;
  __syncthreads();
  bool rowOK = k < neA;
  for (int w = 0; w < 16; ++w) {
    unsigned bits = 0u;
    for (int bi = 0; bi < 32; ++bi) {
      int l = w * 32 + bi;
      float dpv = p0 * pc[l][0] + p1 * pc[l][1] + p2 * pc[l][2];
      float sqp = npk + npn[l] - 2.f * dpv;
      float dp = sqrtf(fmaxf(sqp, 0.f) + 1e-12f);
      float dqv = q0 * qc[l][0] + q1 * qc[l][1] + q2 * qc[l][2];
      float sqq = nqk + nqn[l] - 2.f * dqv;
      float dq = sqrtf(fmaxf(sqq, 0.f) + 1e-12f);
      bool ok = rowOK && (l < neN) && (l != k) && (fabsf(dp - dq) < 0.5f);
      bits |= (ok ? 1u : 0u) << bi;
    }
    geo[((size_t)tn * KK + k) * 16 + w] = bits;
  }
}

// ---------------------------------------------------------------------------
// Kernel 7: 20 power iterations of A v (A = geo * s_k * s_l, factored through
// the bitmask), normalize, then bitonic sort descending -> eig. 128 x 512.
// ---------------------------------------------------------------------------
__global__ void power_sort_kernel(const unsigned* __restrict__ geo,
                                  const float* __restrict__ sbuf,
                                  float* __restrict__ eig) {
  int tn = blockIdx.x;
  int k = threadIdx.x;
  int lane = k & 31, wave = k >> 5;
  float sk = sbuf[(size_t)tn * KK + k];
  unsigned bits[16];
  const unsigned* row = geo + ((size_t)tn * KK + k) * 16;
#pragma unroll
  for (int w = 0; w < 16; ++w) bits[w] = row[w];
  __shared__ float u[KK];
  __shared__ float red[16];
  __shared__ float totN;
  float v = 1.0f / sqrtf((float)KK);
  for (int it = 0; it < 20; ++it) {
    u[k] = sk * v;
    __syncthreads();
    float sum = 0.f;
#pragma unroll
    for (int w = 0; w < 16; ++w) {
      unsigned m = bits[w];
      const float* uw = u + w * 32;
      while (m) { int b = __builtin_ctz(m); sum += uw[b]; m &= m - 1; }
    }
    float wk = sk * sum;
    float sq = wk * wk;
    for (int m2 = 16; m2 >= 1; m2 >>= 1) sq += __shfl_xor(sq, m2, 32);
    if (lane == 0) red[wave] = sq;
    __syncthreads();
    if (k == 0) {
      float tt = 0.f;
      for (int i2 = 0; i2 < 16; ++i2) tt += red[i2];
      totN = tt;
    }
    __syncthreads();
    v = wk / (sqrtf(totN) + 1e-8f);
  }
  u[k] = v;
  __syncthreads();
  for (int ksz = 2; ksz <= KK; ksz <<= 1) {
    for (int j = ksz >> 1; j > 0; j >>= 1) {
      int ixj = k ^ j;
      if (ixj > k) {
        float a = u[k], c = u[ixj];
        bool up = ((k & ksz) == 0);
        if (up ? (a < c) : (a > c)) { u[k] = c; u[ixj] = a; }  // descending
      }
      __syncthreads();
    }
  }
  eig[(size_t)tn * KK + k] = u[k];
}

// ---------------------------------------------------------------------------
// Kernel 8: H = relu(EIG(128x512) @ W1(512x512) + b1), WMMA f32 16x16x4.
// grid (128/16, 512/128), block 256.
// ---------------------------------------------------------------------------
__global__ void h_gemm_kernel(const float* __restrict__ eig,
                              const float* __restrict__ W1,
                              const float* __restrict__ b1,
                              float* __restrict__ Hb) {
  int lane = threadIdx.x & 31, wave = threadIdx.x >> 5;
  int rowTile = blockIdx.x;
  int colBase = blockIdx.y * 128 + wave * 16;
  int l15 = lane & 15;
  int kOff = (lane >> 4) * 2;
  const float* Arow = eig + (size_t)(rowTile * 16 + l15) * KK;
  int ncol = colBase + l15;
  v8f acc = {};
  for (int k0 = 0; k0 < KK; k0 += 4) {
    v2f a = *(const v2f*)(Arow + k0 + kOff);
    v2f bm;
    bm[0] = W1[(size_t)(k0 + kOff) * KK + ncol];
    bm[1] = W1[(size_t)(k0 + kOff + 1) * KK + ncol];
    acc = wmma4(a, bm, acc);
  }
  float bias = b1[ncol];
  int rbase = rowTile * 16 + (lane >> 4) * 8;
#pragma unroll
  for (int r = 0; r < 8; ++r)
    Hb[(size_t)(rbase + r) * KK + ncol] = fmaxf(acc[r] + bias, 0.f);
}

// ---------------------------------------------------------------------------
// Kernel 9: scores = sigmoid(H @ W2 + b2); write scores then targets.
// ---------------------------------------------------------------------------
__global__ void score_kernel(const float* __restrict__ Hb,
                             const float* __restrict__ W2,
                             const float* __restrict__ b2,
                             float* __restrict__ out) {
  int tn = blockIdx.x;
  int tid = threadIdx.x;
  float s = 0.f;
  for (int j = tid; j < KK; j += 256) s += Hb[(size_t)tn * KK + j] * W2[j];
  int lane = tid & 31, wave = tid >> 5;
  for (int m = 16; m >= 1; m >>= 1) s += __shfl_xor(s, m, 32);
  __shared__ float red[8];
  if (lane == 0) red[wave] = s;
  __syncthreads();
  if (tid == 0) {
    float tot = 0.f;
    for (int i = 0; i < 8; ++i) tot += red[i];
    float x = tot + b2[0];
    out[tn] = 1.0f / (1.0f + expf(-x));
    out[TT * 2 + tn] = ((tn & 1) == 0) ? 1.0f : 0.0f;  // targets
  }
}

// ---------------------------------------------------------------------------
extern "C" void kernel_launch(void* const* d_in, const int* in_sizes, int n_in,
                              void* d_out, int out_size, void* d_ws,
                              size_t ws_size, hipStream_t stream) {
  (void)in_sizes; (void)n_in; (void)out_size; (void)ws_size;
  const float* rt = (const float*)d_in[0];
  const float* cent = (const float*)d_in[1];
  const float* attn = (const float*)d_in[2];
  const float* sas = (const float*)d_in[3];
  const int* num_rt = (const int*)d_in[4];
  const int* anc = (const int*)d_in[5];
  const int* pos = (const int*)d_in[6];
  const int* neg = (const int*)d_in[7];
  const float* W_in = (const float*)d_in[8];
  const float* b_in = (const float*)d_in[9];
  const float* W1 = (const float*)d_in[10];
  const float* b1 = (const float*)d_in[11];
  const float* W2 = (const float*)d_in[12];
  const float* b2 = (const float*)d_in[13];
  float* out = (float*)d_out;

  char* ws = (char*)d_ws;
  size_t off = 0;
  auto carve = [&](size_t bytes) -> char* {
    char* p = ws + off;
    off += (bytes + 255) & ~(size_t)255;
    return p;
  };
  int* topIdx = (int*)carve((size_t)BB * KK * 4);
  float* centK = (float*)carve((size_t)BB * KK * 3 * 4);
  float* proj = (float*)carve((size_t)BB * KK * CC * 4);
  float* n2 = (float*)carve((size_t)BB * KK * 4);
  int* corr = (int*)carve((size_t)TT * 2 * KK * 4);
  float* sbuf = (float*)carve((size_t)TT * 2 * KK * 4);
  float* qcb = (float*)carve((size_t)TT * 2 * KK * 3 * 4);
  unsigned* geo = (unsigned*)carve((size_t)TT * 2 * KK * 16 * 4);
  float* eig = (float*)carve((size_t)TT * 2 * KK * 4);
  float* Hb = (float*)carve((size_t)TT * 2 * KK * 4);

  topk_kernel<<<BB, 512, 0, stream>>>(attn, cent, num_rt, sas, topIdx, centK);
  proj_gemm_kernel<<<dim3(KK / 16, CC / 128, BB), 256, 0, stream>>>(
      rt, W_in, b_in, topIdx, proj);
  rownorm_kernel<<<(BB * KK) / 8, 256, 0, stream>>>(proj, n2);
  fd_argmin_kernel<<<dim3(KK / 16, 2, TT), 256, 0, stream>>>(proj, n2, anc,
                                                             pos, neg, corr);
  s_qc_kernel<<<(TT * 2 * KK) / 8, 256, 0, stream>>>(proj, n2, centK, corr,
                                                     anc, pos, neg, sbuf, qcb);
  geo_kernel<<<TT * 2, 512, 0, stream>>>(centK, qcb, num_rt, anc, pos, neg, geo);
  power_sort_kernel<<<TT * 2, 512, 0, stream>>>(geo, sbuf, eig);
  h_gemm_kernel<<<dim3((TT * 2) / 16, KK / 128), 256, 0, stream>>>(eig, W1, b1, Hb);
  score_kernel<<<TT * 2, 256, 0, stream>>>(Hb, W2, b2, out);
}